// BoundaryLoss_39152921870910
// MI455X (gfx1250) — compile-verified
//
#include <hip/hip_runtime.h>
#include <stdint.h>

// BoundaryLoss on MI455X (gfx1250).
// loss == 100 * mean( boundary(sigmoid(logits)) != boundary(targets) )
// where boundary = (sobel_gx^2 + sobel_gy^2 > 0.25), SAME zero padding per
// (B,C) plane. Memory-bound: 134 MB in / 23.3 TB/s ~ 5.8 us floor.
// Data path: async global->LDS tile staging (CDNA5 GLOBAL_LOAD_ASYNC_TO_LDS_B128
// for the aligned interior + _B32 for the x-halo, tracked by ASYNCcnt),
// Sobel stencil out of LDS, deterministic integer mismatch-count reduction.

#define IMG_H 512
#define IMG_W 512
#define TILE 64
#define HALO (TILE + 2)   // 66 rows / 66 halo columns
#define LSTRIDE 72        // words per LDS row; halo cols live at [3..68] so the
                          // 64 interior cols start 16B-aligned (ly*288 + 16)
#define COL0 3            // LDS column of halo coord lxh==0
#define NPLANES 64        // B*C = 16*4
#define NELEM (16LL * 4 * 512 * 512)

__device__ __forceinline__ unsigned lds_offset(const void* p) {
    // Generic pointer to LDS: low 32 bits are the wave-relative LDS byte offset.
    return (unsigned)(uintptr_t)p;
}

__device__ __forceinline__ void async_b32_to_lds(unsigned lds_byte_off,
                                                 const float* gptr) {
    asm volatile("global_load_async_to_lds_b32 %0, %1, off"
                 :
                 : "v"(lds_byte_off), "v"((uint64_t)(uintptr_t)gptr)
                 : "memory");
}

__device__ __forceinline__ void async_b128_to_lds(unsigned lds_byte_off,
                                                  const float* gptr) {
    asm volatile("global_load_async_to_lds_b128 %0, %1, off"
                 :
                 : "v"(lds_byte_off), "v"((uint64_t)(uintptr_t)gptr)
                 : "memory");
}

__device__ __forceinline__ void wait_asynccnt0() {
    asm volatile("s_wait_asynccnt 0" ::: "memory");
}

__global__ __launch_bounds__(256) void boundary_mismatch_kernel(
    const float* __restrict__ logits,
    const float* __restrict__ targets,
    unsigned* __restrict__ g_count) {
    __shared__ float sP[HALO * LSTRIDE];  // sigmoid(logits) tile (+halo)
    __shared__ float sT[HALO * LSTRIDE];  // targets tile (+halo)
    __shared__ unsigned s_cnt;

    const int tid = threadIdx.x;
    const int tx0 = blockIdx.x * TILE;
    const int ty0 = blockIdx.y * TILE;
    const size_t pbase = (size_t)blockIdx.z * (IMG_H * IMG_W);

    if (tid == 0) s_cnt = 0;
    // Zero targets tile: halo outside the image must read 0 (SAME padding),
    // and skipped (OOB) async loads leave LDS untouched. sP needs no pre-zero:
    // the sigmoid pass rewrites every slot.
    for (int i = tid; i < HALO * LSTRIDE; i += 256) sT[i] = 0.0f;
    __syncthreads();

    const unsigned baseP = lds_offset(sP);
    const unsigned baseT = lds_offset(sT);

    // ---- Bulk interior: 66 rows x 16 quads of 4 floats, 16B-aligned both in
    // global (tx0 % 64 == 0) and LDS (ly*288 + 16 + 16q). One b128 per task.
    for (int t = tid; t < HALO * 16; t += 256) {
        const int ly = t >> 4;        // 0..65
        const int q = t & 15;         // quad within the 64 interior columns
        const int gy = ty0 - 1 + ly;
        if (gy >= 0 && gy < IMG_H) {
            const size_t goff = pbase + (size_t)gy * IMG_W + tx0 + q * 4;
            const unsigned loff = (unsigned)(ly * LSTRIDE + (COL0 + 1) + q * 4) * 4u;
            async_b128_to_lds(baseP + loff, logits + goff);
            async_b128_to_lds(baseT + loff, targets + goff);
        }
    }
    // ---- x-halo columns: gx = tx0-1 and tx0+64 (one b32 per row per side).
    for (int t = tid; t < HALO * 2; t += 256) {
        const int ly = t >> 1;
        const int side = t & 1;                       // 0: left, 1: right
        const int gy = ty0 - 1 + ly;
        const int gx = side ? (tx0 + TILE) : (tx0 - 1);
        if (gy >= 0 && gy < IMG_H && gx >= 0 && gx < IMG_W) {
            const size_t goff = pbase + (size_t)gy * IMG_W + gx;
            const int lc = side ? (COL0 + HALO - 1) : COL0;
            const unsigned loff = (unsigned)(ly * LSTRIDE + lc) * 4u;
            async_b32_to_lds(baseP + loff, logits + goff);
            async_b32_to_lds(baseT + loff, targets + goff);
        }
    }
    wait_asynccnt0();
    __syncthreads();

    // Sigmoid in place on the logits tile; force OOB halo slots to 0.0
    // (padding applies to probs, NOT sigmoid(0)=0.5).
    for (int i = tid; i < HALO * HALO; i += 256) {
        const int ly = i / HALO;
        const int lxh = i - ly * HALO;                // halo coord 0..65
        const int gy = ty0 - 1 + ly;
        const int gx = tx0 - 1 + lxh;
        const bool inb = (gy >= 0) & (gy < IMG_H) & (gx >= 0) & (gx < IMG_W);
        const int li = ly * LSTRIDE + lxh + COL0;
        const float v = sP[li];
        sP[li] = inb ? __frcp_rn(1.0f + __expf(-v)) : 0.0f;
    }
    __syncthreads();

    // Each thread: one column of a 16-row strip, rolling 3x3 register window.
    // Every LDS load has 32 lanes at consecutive addresses -> conflict-free.
    const int lx = (tid & 63) + 1;            // halo coord 1..64
    const int ly0 = ((tid >> 6) << 4) + 1;    // 1, 17, 33, 49
    const int cbase = lx - 1 + COL0;          // LDS column of window's left edge

    float P[3][3], T[3][3];
#pragma unroll
    for (int r = 0; r < 2; ++r)
#pragma unroll
        for (int c = 0; c < 3; ++c) {
            P[r][c] = sP[(ly0 - 1 + r) * LSTRIDE + cbase + c];
            T[r][c] = sT[(ly0 - 1 + r) * LSTRIDE + cbase + c];
        }

    unsigned cnt = 0;
#pragma unroll
    for (int k = 0; k < 16; ++k) {
        const int y = ly0 + k;
#pragma unroll
        for (int c = 0; c < 3; ++c) {
            P[2][c] = sP[(y + 1) * LSTRIDE + cbase + c];
            T[2][c] = sT[(y + 1) * LSTRIDE + cbase + c];
        }
        // Sobel gx/gy; threshold on magnitude^2 (mag>0.5 <=> mag^2>0.25)
        const float pgx = (P[0][2] - P[0][0]) + 2.0f * (P[1][2] - P[1][0]) +
                          (P[2][2] - P[2][0]);
        const float pgy = (P[2][0] + 2.0f * P[2][1] + P[2][2]) -
                          (P[0][0] + 2.0f * P[0][1] + P[0][2]);
        const float tgx = (T[0][2] - T[0][0]) + 2.0f * (T[1][2] - T[1][0]) +
                          (T[2][2] - T[2][0]);
        const float tgy = (T[2][0] + 2.0f * T[2][1] + T[2][2]) -
                          (T[0][0] + 2.0f * T[0][1] + T[0][2]);
        const bool pb = (pgx * pgx + pgy * pgy) > 0.25f;
        const bool tb = (tgx * tgx + tgy * tgy) > 0.25f;
        cnt += (unsigned)(pb != tb);
#pragma unroll
        for (int c = 0; c < 3; ++c) {
            P[0][c] = P[1][c]; P[1][c] = P[2][c];
            T[0][c] = T[1][c]; T[1][c] = T[2][c];
        }
    }

    atomicAdd(&s_cnt, cnt);
    __syncthreads();
    if (tid == 0 && s_cnt != 0) atomicAdd(g_count, s_cnt);
}

__global__ void init_count_kernel(unsigned* c) { *c = 0u; }

__global__ void finalize_kernel(const unsigned* __restrict__ c,
                                float* __restrict__ out) {
    out[0] = 100.0f * ((float)(*c) / (float)NELEM);
}

extern "C" void kernel_launch(void* const* d_in, const int* in_sizes, int n_in,
                              void* d_out, int out_size, void* d_ws,
                              size_t ws_size, hipStream_t stream) {
    const float* logits = (const float*)d_in[0];
    const float* targets = (const float*)d_in[1];
    unsigned* g_count = (unsigned*)d_ws;
    float* out = (float*)d_out;

    init_count_kernel<<<1, 1, 0, stream>>>(g_count);

    dim3 grid(IMG_W / TILE, IMG_H / TILE, NPLANES);  // 8 x 8 x 64
    boundary_mismatch_kernel<<<grid, 256, 0, stream>>>(logits, targets, g_count);

    finalize_kernel<<<1, 1, 0, stream>>>(g_count, out);
}